// QuantizedKVCache_13597866459500
// MI455X (gfx1250) — compile-verified
//
#include <hip/hip_runtime.h>
#include <stdint.h>

// QuantizedKVCache update+dequant for MI455X (gfx1250).
// out[kv][b,h,t,d] = (sp <= t < sp+S) ? val[b,h,t-sp,d]
//                                     : (cache[b,t,h,d] - zp[b,t,h]) * scale[b,t,h]
// Pure streaming op (~324 MiB traffic -> ~14us at 23.3 TB/s). No matmul survives
// dead-code elimination, so no WMMA; the CDNA5-specific path used is the async
// global->LDS DMA (global_load_async_to_lds_b128 / s_wait_asynccnt) plus
// non-temporal 128-bit stores for the 256 MiB write-once output.

typedef __attribute__((ext_vector_type(4))) float f32x4;
typedef __attribute__((ext_vector_type(4))) int   i32x4;

namespace {
constexpr int B    = 4;
constexpr int H    = 32;
constexpr int S    = 512;
constexpr int D    = 128;
constexpr int SMAX = 2048;
constexpr int HD   = H * D;                         // 4096 bytes of int8 per (b,t) slab
constexpr size_t OUT_ELEMS = (size_t)B * H * SMAX * D; // 33,554,432 floats per tensor
}

__global__ __launch_bounds__(256) void kvcache_dequant_kernel(
    const int*    __restrict__ input_pos,
    const float*  __restrict__ k_val,  const float*  __restrict__ v_val,
    const int8_t* __restrict__ k_cache, const int8_t* __restrict__ v_cache,
    const float*  __restrict__ k_scales, const float* __restrict__ v_scales,
    const int*    __restrict__ k_zp,   const int*    __restrict__ v_zp,
    float*        __restrict__ out)
{
    __shared__ __align__(16) int8_t lds_q[HD];      // one 4 KiB int8 slab per block

    const int tid = threadIdx.x;
    const int bid = blockIdx.x;
    const int kv  = bid >> 13;                      // / (B*SMAX) == /8192
    const int rem = bid & 8191;
    const int b   = rem >> 11;                      // / SMAX
    const int t   = rem & (SMAX - 1);

    int sp = input_pos[0];
    sp = sp < 0 ? 0 : (sp > SMAX - S ? SMAX - S : sp);

    const float*  val    = kv ? v_val    : k_val;
    const int8_t* cache  = kv ? v_cache  : k_cache;
    const float*  scales = kv ? v_scales : k_scales;
    const int*    zps    = kv ? v_zp     : k_zp;
    float* o = out + (size_t)kv * OUT_ELEMS;

    const int h  = tid >> 3;                        // 8 threads per head row
    const int d0 = (tid & 7) << 4;                  // 16 floats per thread

    float* dst = o + ((size_t)((b * H + h) * SMAX + t) * D + d0);
    f32x4* d4  = (f32x4*)dst;

    if (t >= sp && t < sp + S) {
        // Fresh region: exact fp passthrough from [B,H,S,D] input.
        const f32x4* src =
            (const f32x4*)(val + ((size_t)((b * H + h) * S + (t - sp)) * D + d0));
#pragma unroll
        for (int i = 0; i < 4; ++i) {
            f32x4 v4 = __builtin_nontemporal_load(src + i);
            __builtin_nontemporal_store(v4, d4 + i);
        }
    } else {
        // Stale region: async-DMA the contiguous 4 KiB [H,D] int8 slab into LDS.
        // Each lane transfers and later consumes its own 16 bytes, so a
        // per-wave s_wait_asynccnt 0 is sufficient (no barrier needed).
        const int slab = (b * SMAX + t) * HD;
        const int8_t* gsrc = cache + slab + tid * 16;
        unsigned lds_off = (unsigned)(uintptr_t)&lds_q[tid * 16];
        asm volatile("global_load_async_to_lds_b128 %0, %1, off"
                     :: "v"(lds_off), "v"(gsrc)
                     : "memory");
        asm volatile("s_wait_asynccnt 0x0" ::: "memory");

        i32x4 qw = *(const i32x4*)&lds_q[tid * 16];

        const int   sidx  = (b * SMAX + t) * H + h;
        const float scale = scales[sidx];
        const int   zp    = zps[sidx];

#pragma unroll
        for (int i = 0; i < 4; ++i) {
            const int w = qw[i];
            f32x4 f;
#pragma unroll
            for (int j = 0; j < 4; ++j) {
                const int q = (int)(int8_t)((unsigned)w >> (8 * j));
                // (q - zp) is exact in int, matching the reference's
                // (q.f32 - zp.f32) * scale with a single rounding on the mul.
                f[j] = (float)(q - zp) * scale;
            }
            __builtin_nontemporal_store(f, d4 + i);
        }
    }
}

extern "C" void kernel_launch(void* const* d_in, const int* in_sizes, int n_in,
                              void* d_out, int out_size, void* d_ws, size_t ws_size,
                              hipStream_t stream) {
    (void)in_sizes; (void)n_in; (void)out_size; (void)d_ws; (void)ws_size;

    const int*    input_pos = (const int*)   d_in[0];
    const float*  k_val     = (const float*) d_in[1];
    const float*  v_val     = (const float*) d_in[2];
    const int8_t* k_cache   = (const int8_t*)d_in[3];
    const int8_t* v_cache   = (const int8_t*)d_in[4];
    const float*  k_scales  = (const float*) d_in[5];
    const float*  v_scales  = (const float*) d_in[6];
    const int*    k_zp      = (const int*)   d_in[7];
    const int*    v_zp      = (const int*)   d_in[8];
    float*        out       = (float*)       d_out;

    const int blocks = 2 * B * SMAX;  // 16384 blocks: (k|v) x b x t
    kvcache_dequant_kernel<<<blocks, 256, 0, stream>>>(
        input_pos, k_val, v_val, k_cache, v_cache,
        k_scales, v_scales, k_zp, v_zp, out);
}